// EncoderBlock_9079560864617
// MI455X (gfx1250) — compile-verified
//
#include <hip/hip_runtime.h>
#include <math.h>

// ---------------- geometry ----------------
#define Bz     16
#define Dm     2048
#define NH     32
#define NKV    8
#define HD     64
#define GRP    4
#define MAXSEQ 4096
#define LASTT  4095      // start_pos, S=1
#define HID    5632
#define KC     256       // split-K chunk
#define TCH    256       // attention t-chunk
#define NCH    16        // MAXSEQ / TCH

typedef float v2f __attribute__((ext_vector_type(2)));
typedef float v8f __attribute__((ext_vector_type(8)));

// ---------------- ws layout (floats) ----------------
#define OFF_XN    0u
#define OFF_QB    32768u
#define OFF_KF    65536u
#define OFF_VF    73728u
#define OFF_ATT   81920u
#define OFF_H     114688u
#define OFF_HN    147456u
#define OFF_U     180224u
#define OFF_T1    270336u
#define OFF_PART  303104u      // up to 22*16*2048 = 720896 floats
#define OFF_APART 1024000u     // 16*8*16*4*66 = 540672 floats

// ---------------- RMSNorm ----------------
__global__ __launch_bounds__(256) void rmsnorm_k(const float* __restrict__ x,
                                                 const float* __restrict__ w,
                                                 float* __restrict__ out) {
  const int b = blockIdx.x, tid = threadIdx.x;
  const float* xr = x + (size_t)b * Dm;
  float ss = 0.f;
  for (int i = tid; i < Dm; i += 256) { float v = xr[i]; ss += v * v; }
  __shared__ float red[256];
  red[tid] = ss; __syncthreads();
  for (int s = 128; s > 0; s >>= 1) { if (tid < s) red[tid] += red[tid + s]; __syncthreads(); }
  const float r = rsqrtf(red[0] / (float)Dm + 1e-6f);
  for (int i = tid; i < Dm; i += 256) out[(size_t)b * Dm + i] = w[i] * xr[i] * r;
}

// ---------------- WMMA fp32 GEMM: C[16,N] = A[16,K] * W[N,K]^T, split-K partials ----------------
// One wave computes a 16x32 tile over one 256-wide K chunk.
// Pointer-induction inner loop: all loads are base + constant immediate offset.
// Partial layout (per chunk of 16*N floats): tile x at x*512; within tile,
// element (j, r, lane) at (j*8 + r)*32 + lane  ->  m = r + (lane>>4)*8,
// n = x*32 + j*16 + (lane&15).  Epilogue = one s_clause of 16 stores.
__global__ __launch_bounds__(32) void gemm16_wmma(const float* __restrict__ A,
                                                  const float* __restrict__ W,
                                                  float* __restrict__ part,
                                                  int N, int K) {
  const int lane = threadIdx.x;
  const int n0   = blockIdx.x * 32;
  const int kc   = blockIdx.y;
  const int m    = lane & 15;
  const int koff = (lane >> 4) * 2;             // A/B frag: lanes 0-15 hold K+{0,1}, 16-31 hold K+{2,3}
  const int kbeg = kc * KC;

  v8f acc0 = {0.f,0.f,0.f,0.f,0.f,0.f,0.f,0.f};
  v8f acc1 = {0.f,0.f,0.f,0.f,0.f,0.f,0.f,0.f};

  // per-lane stream pointers with koff folded in; inner offsets are constants
  const float* pa = A + (size_t)m * K            + kbeg + koff;
  const float* p0 = W + (size_t)(n0 + m) * K      + kbeg + koff;
  const float* p1 = W + (size_t)(n0 + 16 + m) * K + kbeg + koff;

  for (int kg = 0; kg < KC; kg += 64) {
    __builtin_prefetch(p0 + 320, 0, 0);         // global_prefetch_b8, imm offset, GL2 only
    __builtin_prefetch(p1 + 320, 0, 0);
#pragma unroll
    for (int k0 = 0; k0 < 64; k0 += 4) {        // constant offsets 0..240 bytes
      v2f a  = *(const v2f*)(pa + k0);
      v2f b0 = *(const v2f*)(p0 + k0);
      v2f b1 = *(const v2f*)(p1 + k0);
      acc0 = __builtin_amdgcn_wmma_f32_16x16x4_f32(false, a, false, b0, (short)0, acc0, false, false);
      acc1 = __builtin_amdgcn_wmma_f32_16x16x4_f32(false, a, false, b1, (short)0, acc1, false, false);
    }
    pa += 64; p0 += 64; p1 += 64;               // 3 pointer bumps per 16 WMMA steps
  }

  float* __restrict__ p = part + (size_t)kc * 16 * N + (size_t)blockIdx.x * 512 + lane;
#pragma unroll
  for (int r = 0; r < 8; ++r) {
    p[r * 32]       = acc0[r];
    p[(8 + r) * 32] = acc1[r];
  }
}

// sum split-K partials (+ optional residual base, + optional SiLU) — deterministic
__global__ __launch_bounds__(256) void gemm_reduce(const float* __restrict__ part,
                                                   int nch, int N,
                                                   const float* __restrict__ base,
                                                   float* __restrict__ out, int act) {
  const int i = blockIdx.x * 256 + threadIdx.x;  // index in wave-tile partial layout
  if (i >= 16 * N) return;
  float s = 0.f;
  for (int c = 0; c < nch; ++c) s += part[(size_t)c * 16 * N + i];
  // unscramble (tile, j, r, lane) -> (m, n)
  const int tile = i >> 9, rem = i & 511;
  const int j = rem >> 8, r = (rem >> 5) & 7, lane = rem & 31;
  const int mr = r + (lane >> 4) * 8;
  const int n  = tile * 32 + j * 16 + (lane & 15);
  const size_t o = (size_t)mr * N + n;
  if (base) s += base[o];
  if (act) s = s / (1.f + __expf(-s));          // SiLU
  out[o] = s;
}

// ---------------- RoPE on fresh q,k,v (reference applies it to v too) ----------------
__global__ __launch_bounds__(32) void rope_k(float* __restrict__ q, float* __restrict__ k,
                                             float* __restrict__ v,
                                             const float* __restrict__ cosp,
                                             const float* __restrict__ sinp) {
  const int gid = blockIdx.x;                   // b*48 + slot
  const int slot = gid % 48, b = gid / 48;
  float* p; int head, nh;
  if (slot < 32)      { p = q; head = slot;      nh = NH; }
  else if (slot < 40) { p = k; head = slot - 32; nh = NKV; }
  else                { p = v; head = slot - 40; nh = NKV; }
  const int i = threadIdx.x;                    // pair index 0..31
  const float c = cosp[i], s = sinp[i];
  const size_t base = ((size_t)b * nh + head) * HD;
  const float x0 = p[base + 2 * i], x1 = p[base + 2 * i + 1];
  p[base + 2 * i]     = x0 * c - x1 * s;
  p[base + 2 * i + 1] = x0 * s + x1 * c;
}

// ---------------- flash-decode partial: one wave per (b, kv-head, 256-t chunk) ----------------
__global__ __launch_bounds__(32) void attn_partial(const float* __restrict__ qb,
                                                   const float* __restrict__ kf,
                                                   const float* __restrict__ vf,
                                                   const float* __restrict__ kcache,
                                                   const float* __restrict__ vcache,
                                                   float* __restrict__ apart) {
  const int gid  = blockIdx.x;
  const int ch   = gid & 15;
  const int kvh  = (gid >> 4) & 7;
  const int b    = gid >> 7;
  const int lane = threadIdx.x;
  const int m    = lane & 15;
  const int koff = (lane >> 4) * 2;
  const int tbase = ch * TCH;
  const bool last_ch = (ch == NCH - 1);

  __shared__ float sp[GRP * TCH];               // scores -> probs

  // preload Q A-fragments (rows 0..3 = group heads, rows 4..15 zero-padded)
  v2f aq[16];
  const float* qrow = qb + ((size_t)b * NH + kvh * GRP + (m & 3)) * HD + koff;
#pragma unroll
  for (int k0 = 0; k0 < 16; ++k0) {
    v2f a = *(const v2f*)(qrow + k0 * 4);       // constant offsets
    if (m >= GRP) { a.x = 0.f; a.y = 0.f; }
    aq[k0] = a;
  }

  // QK^T over 16 key tiles of 16, WMMA f32 16x16x4 (K=HD in 16 steps)
  const float* kfrow = kf + ((size_t)b * NKV + kvh) * HD;
  for (int nt = 0; nt < 16; ++nt) {
    const int t = tbase + nt * 16 + m;          // this lane's key column
    const float* kp = ((last_ch && t == LASTT)
        ? kfrow
        : (kcache + (((size_t)b * MAXSEQ + t) * NKV + kvh) * HD)) + koff;
    v8f acc = {0.f,0.f,0.f,0.f,0.f,0.f,0.f,0.f};
#pragma unroll
    for (int k0 = 0; k0 < 16; ++k0) {
      v2f bf = *(const v2f*)(kp + k0 * 4);      // constant offsets
      acc = __builtin_amdgcn_wmma_f32_16x16x4_f32(false, aq[k0], false, bf, (short)0, acc, false, false);
    }
    if (lane < 16) {
#pragma unroll
      for (int r = 0; r < GRP; ++r) sp[r * TCH + nt * 16 + lane] = acc[r] * 0.125f; // 1/sqrt(64)
    }
  }
  __syncthreads();

  // per-group online-softmax partial over this chunk
  float mg[GRP], lg[GRP];
#pragma unroll
  for (int g = 0; g < GRP; ++g) {
    float mx = -3.4e38f;
    for (int j = 0; j < TCH / 32; ++j) mx = fmaxf(mx, sp[g * TCH + lane + j * 32]);
    for (int o = 16; o > 0; o >>= 1) mx = fmaxf(mx, __shfl_xor(mx, o, 32));
    float sum = 0.f;
    for (int j = 0; j < TCH / 32; ++j) {
      const int t = lane + j * 32;
      const float e = __expf(sp[g * TCH + t] - mx);
      sp[g * TCH + t] = e;
      sum += e;
    }
    for (int o = 16; o > 0; o >>= 1) sum += __shfl_xor(sum, o, 32);
    mg[g] = mx; lg[g] = sum;
  }
  __syncthreads();

  // P*V: lane owns dims {2*lane, 2*lane+1}; V float2 loads are fully coalesced.
  // Clean streaming loop; the single fresh row (t==4095) handled as a tail.
  const int d0 = lane * 2;
  float a0[GRP] = {0,0,0,0}, a1[GRP] = {0,0,0,0};
  const float* vrow = vcache + (((size_t)b * MAXSEQ + tbase) * NKV + kvh) * HD + d0;
  const int tlim = last_ch ? TCH - 1 : TCH;
  for (int tt = 0; tt < tlim; ++tt) {
    const v2f v = *(const v2f*)(vrow + (size_t)tt * (NKV * HD));
#pragma unroll
    for (int g = 0; g < GRP; ++g) {
      const float p = sp[g * TCH + tt];         // LDS broadcast, conflict-free
      a0[g] += p * v.x;
      a1[g] += p * v.y;
    }
  }
  if (last_ch) {                                // fresh (roped) V at t = 4095
    const v2f v = *(const v2f*)(vf + ((size_t)b * NKV + kvh) * HD + d0);
#pragma unroll
    for (int g = 0; g < GRP; ++g) {
      const float p = sp[g * TCH + (TCH - 1)];
      a0[g] += p * v.x;
      a1[g] += p * v.y;
    }
  }

  float* pb = apart + ((((size_t)b * NKV + kvh) * NCH + ch) * GRP) * 66;
#pragma unroll
  for (int g = 0; g < GRP; ++g) {
    float* pg = pb + (size_t)g * 66;
    if (lane == 0) { pg[0] = mg[g]; pg[1] = lg[g]; }
    pg[2 + d0]     = a0[g];
    pg[2 + d0 + 1] = a1[g];
  }
}

// combine chunk partials (log-sum-exp merge), write attn activations [16, 2048]
__global__ __launch_bounds__(32) void attn_reduce(const float* __restrict__ apart,
                                                  float* __restrict__ attnb) {
  const int hidx = blockIdx.x;                  // b*32 + kvh*4 + g
  const int g = hidx & 3, kvh = (hidx >> 2) & 7, b = hidx >> 5;
  const int lane = threadIdx.x, d0 = lane * 2;
  const float* hb = apart + (((size_t)b * NKV + kvh) * NCH * GRP + g) * 66;
  float M = -3.4e38f;
  for (int ch = 0; ch < NCH; ++ch) M = fmaxf(M, hb[(size_t)ch * GRP * 66]);
  float L = 0.f, a0 = 0.f, a1 = 0.f;
  for (int ch = 0; ch < NCH; ++ch) {
    const float* pg = hb + (size_t)ch * GRP * 66;
    const float w = __expf(pg[0] - M);
    L  += pg[1] * w;
    a0 += pg[2 + d0] * w;
    a1 += pg[3 + d0] * w;
  }
  const float inv = 1.f / L;
  const size_t o = (size_t)b * Dm + (size_t)(kvh * GRP + g) * HD + d0;
  attnb[o]     = a0 * inv;
  attnb[o + 1] = a1 * inv;
}

// ---------------- launcher ----------------
extern "C" void kernel_launch(void* const* d_in, const int* in_sizes, int n_in,
                              void* d_out, int out_size, void* d_ws, size_t ws_size,
                              hipStream_t stream) {
  const float* x   = (const float*)d_in[0];
  const float* fc  = (const float*)d_in[1];
  const float* fs  = (const float*)d_in[2];
  const float* ck  = (const float*)d_in[3];
  const float* cv  = (const float*)d_in[4];
  const float* wq  = (const float*)d_in[5];
  const float* wk  = (const float*)d_in[6];
  const float* wv  = (const float*)d_in[7];
  const float* wo  = (const float*)d_in[8];
  const float* w1  = (const float*)d_in[9];
  const float* w2  = (const float*)d_in[10];
  const float* w3  = (const float*)d_in[11];
  const float* anw = (const float*)d_in[12];
  const float* fnw = (const float*)d_in[13];
  float* out = (float*)d_out;
  float* ws  = (float*)d_ws;

  const int red2048 = (16 * 2048 + 255) / 256;
  const int red512  = (16 * 512  + 255) / 256;
  const int red5632 = (16 * 5632 + 255) / 256;

  // 1) xn = rmsnorm(x) * attn_norm_w
  rmsnorm_k<<<Bz, 256, 0, stream>>>(x, anw, ws + OFF_XN);

  // 2) Q/K/V projections (split-K WMMA + deterministic reduce)
  gemm16_wmma<<<dim3(2048/32, 2048/KC), 32, 0, stream>>>(ws + OFF_XN, wq, ws + OFF_PART, 2048, 2048);
  gemm_reduce<<<red2048, 256, 0, stream>>>(ws + OFF_PART, 2048/KC, 2048, nullptr, ws + OFF_QB, 0);
  gemm16_wmma<<<dim3(512/32, 2048/KC), 32, 0, stream>>>(ws + OFF_XN, wk, ws + OFF_PART, 512, 2048);
  gemm_reduce<<<red512, 256, 0, stream>>>(ws + OFF_PART, 2048/KC, 512, nullptr, ws + OFF_KF, 0);
  gemm16_wmma<<<dim3(512/32, 2048/KC), 32, 0, stream>>>(ws + OFF_XN, wv, ws + OFF_PART, 512, 2048);
  gemm_reduce<<<red512, 256, 0, stream>>>(ws + OFF_PART, 2048/KC, 512, nullptr, ws + OFF_VF, 0);

  // 3) RoPE on q, k, v
  rope_k<<<Bz * 48, 32, 0, stream>>>(ws + OFF_QB, ws + OFF_KF, ws + OFF_VF, fc, fs);

  // 4) flash-decode attention over T=4096 (cache rows 0..4094 + fresh row 4095)
  attn_partial<<<Bz * NKV * NCH, 32, 0, stream>>>(ws + OFF_QB, ws + OFF_KF, ws + OFF_VF,
                                                  ck, cv, ws + OFF_APART);
  attn_reduce<<<Bz * NH, 32, 0, stream>>>(ws + OFF_APART, ws + OFF_ATT);

  // 5) h = x + attn @ wo^T
  gemm16_wmma<<<dim3(2048/32, 2048/KC), 32, 0, stream>>>(ws + OFF_ATT, wo, ws + OFF_PART, 2048, 2048);
  gemm_reduce<<<red2048, 256, 0, stream>>>(ws + OFF_PART, 2048/KC, 2048, x, ws + OFF_H, 0);

  // 6) hn = rmsnorm(h) * ffn_norm_w
  rmsnorm_k<<<Bz, 256, 0, stream>>>(ws + OFF_H, fnw, ws + OFF_HN);

  // 7) u = silu(hn @ w1^T)
  gemm16_wmma<<<dim3(5632/32, 2048/KC), 32, 0, stream>>>(ws + OFF_HN, w1, ws + OFF_PART, 5632, 2048);
  gemm_reduce<<<red5632, 256, 0, stream>>>(ws + OFF_PART, 2048/KC, 5632, nullptr, ws + OFF_U, 1);

  // 8) t1 = h + u @ w2^T   (K = 5632 -> 22 chunks)
  gemm16_wmma<<<dim3(2048/32, 5632/KC), 32, 0, stream>>>(ws + OFF_U, w2, ws + OFF_PART, 2048, 5632);
  gemm_reduce<<<red2048, 256, 0, stream>>>(ws + OFF_PART, 5632/KC, 2048, ws + OFF_H, ws + OFF_T1, 0);

  // 9) out = t1 + hn @ w3^T
  gemm16_wmma<<<dim3(2048/32, 2048/KC), 32, 0, stream>>>(ws + OFF_HN, w3, ws + OFF_PART, 2048, 2048);
  gemm_reduce<<<red2048, 256, 0, stream>>>(ws + OFF_PART, 2048/KC, 2048, ws + OFF_T1, out, 0);
}